// ClauseEnhancer_70660801954611
// MI455X (gfx1250) — compile-verified
//
#include <hip/hip_runtime.h>
#include <stdint.h>

// ---------------------------------------------------------------------------
// ClauseEnhancer: out[:, idx] = signs * softmax(inputs[:, idx] * signs) * w
//
// Roofline: write 400MB (full [1e6 x 100] f32 output) + read ~192MB (3 gathered
// columns, 3 distinct 64B lines per 400B row) => ~592MB @ 23.3TB/s ~= 25us.
// Pure streaming-store problem; WMMA irrelevant (0.03 flop/byte).
//
// gfx1250 paths used:
//  * global_load_async_to_lds_b32 gather (ASYNCcnt) + s_wait_asynccnt
//  * non-temporal b128 stores (output stream 400MB >> 192MB L2)
//  * wave32 sizing: 800-thread blocks = 25 waves, so each lane's float4-chunk
//    id (tid % 25) is loop-invariant -> literal-match masks hoist out of the
//    hot store loop.
// ---------------------------------------------------------------------------

#define USE_ASYNC_GATHER 1

typedef float v4f __attribute__((ext_vector_type(4)));

constexpr int P    = 100;      // predicates per row (reference fixed)
constexpr int PV4  = P / 4;    // 25 float4 chunks per row
constexpr int TPB  = 25 * 32;  // 800 threads = 25 wave32s; rows per block = 800
constexpr int ROWS = TPB;      // one row per thread in Phase A
constexpr int MAXL = 4;        // supports up to 4 literals (reference uses 3)

__global__ __launch_bounds__(TPB)
void clause_enhancer_kernel(const float* __restrict__ inputs,
                            const float* __restrict__ signs,
                            const float* __restrict__ weight,
                            const int*   __restrict__ lit_idx,
                            float* __restrict__ out,
                            int B, int L)
{
    __shared__ float s_delta[ROWS][MAXL];  // per-row scatter values (12.5 KB)
    __shared__ float s_gath [ROWS][MAXL];  // async-gather staging   (12.5 KB)
    __shared__ int   s_idx[MAXL];
    __shared__ float s_sgn[MAXL];
    __shared__ float s_w;

    const int tid  = threadIdx.x;
    const int base = blockIdx.x * ROWS;

    if (tid < MAXL) {
        // sentinel index: (0x40000000 >> 2) never equals a chunk id 0..24
        s_idx[tid] = (tid < L) ? lit_idx[tid] : 0x40000000;
        s_sgn[tid] = (tid < L) ? signs[tid]   : 0.0f;
    }
    if (tid == 0) s_w = weight[0];
    __syncthreads();

    // ---- Phase A: one thread per row -> softmax deltas into LDS -----------
    const int row = base + tid;
    if (row < B) {
#if USE_ASYNC_GATHER
        for (int l = 0; l < L; ++l) {
            uint64_t ga = (uint64_t)(uintptr_t)(inputs + (size_t)row * P + s_idx[l]);
            // generic LDS pointer: low 32 bits are the WG-relative LDS offset
            uint32_t la = (uint32_t)(uintptr_t)(&s_gath[tid][l]);
            asm volatile("global_load_async_to_lds_b32 %0, %1, off"
                         :: "v"(la), "v"(ga) : "memory");
        }
        asm volatile("s_wait_asynccnt 0" ::: "memory");
#else
        for (int l = 0; l < L; ++l)
            s_gath[tid][l] =
                __builtin_nontemporal_load(inputs + (size_t)row * P + s_idx[l]);
#endif
        float x[MAXL], e[MAXL];
        for (int l = 0; l < MAXL; ++l) { x[l] = 0.0f; e[l] = 0.0f; }
        float m = -3.4e38f;
        for (int l = 0; l < L; ++l) {
            x[l] = s_gath[tid][l] * s_sgn[l];
            m = fmaxf(m, x[l]);
        }
        float sum = 0.0f;
        for (int l = 0; l < L; ++l) { e[l] = __expf(x[l] - m); sum += e[l]; }
        const float inv = s_w / sum;
        for (int l = 0; l < MAXL; ++l)
            s_delta[tid][l] = (l < L) ? s_sgn[l] * e[l] * inv : 0.0f;
    }
    __syncthreads();

    // ---- Phase B: coalesced NT float4 streaming of 800 rows ---------------
    // chunk id is LOOP-INVARIANT per lane: tid = rl0*25 + chunk.
    const int rl0   = tid / PV4;          // 0..31
    const int chunk = tid - rl0 * PV4;    // 0..24, constant for this lane

    // loop-invariant literal-match predicates (hoisted once)
    bool m0[MAXL], k0[MAXL], k1[MAXL], k2[MAXL], k3[MAXL];
    #pragma unroll
    for (int l = 0; l < MAXL; ++l) {
        const int lc = s_idx[l] >> 2;     // chunk holding literal l
        const int lk = s_idx[l] & 3;      // component within the chunk
        m0[l] = (lc == chunk);
        k0[l] = m0[l] & (lk == 0);
        k1[l] = m0[l] & (lk == 1);
        k2[l] = m0[l] & (lk == 2);
        k3[l] = m0[l] & (lk == 3);
    }
    const bool hit = m0[0] | m0[1] | m0[2] | m0[3];

    #pragma unroll 1
    for (int it = 0; it < PV4; ++it) {
        const int row_local = rl0 + it * (TPB / PV4);   // +32 per iteration
        const int r = base + row_local;
        if (r < B) {
            float a0 = 0.0f, a1 = 0.0f, a2 = 0.0f, a3 = 0.0f;
            if (hit) {
                #pragma unroll
                for (int l = 0; l < MAXL; ++l) {
                    const float dv = s_delta[row_local][l];
                    a0 = k0[l] ? dv : a0;
                    a1 = k1[l] ? dv : a1;
                    a2 = k2[l] ? dv : a2;
                    a3 = k3[l] ? dv : a3;
                }
            }
            v4f v = {a0, a1, a2, a3};
            __builtin_nontemporal_store(
                v, (v4f*)(out + (size_t)r * P + chunk * 4));
        }
    }
}

extern "C" void kernel_launch(void* const* d_in, const int* in_sizes, int n_in,
                              void* d_out, int out_size, void* d_ws, size_t ws_size,
                              hipStream_t stream) {
    const float* inputs = (const float*)d_in[0];
    const float* signs  = (const float*)d_in[1];
    const float* w      = (const float*)d_in[2];
    const int*   lit    = (const int*)d_in[3];
    float*       out    = (float*)d_out;

    int L = in_sizes[1];
    if (L > MAXL) L = MAXL;
    const int B = in_sizes[0] / P;

    const int grid = (B + ROWS - 1) / ROWS;
    clause_enhancer_kernel<<<grid, TPB, 0, stream>>>(inputs, signs, w, lit, out, B, L);
}